// SOTA_Temporal_PGAT_Enhanced_4002909520448
// MI455X (gfx1250) — compile-verified
//
#include <hip/hip_runtime.h>
#include <hip/hip_bf16.h>
#include <math.h>

// ---------------- problem constants ----------------
#define Bc   8
#define Lc   128
#define Dc   512
#define Hc   8
#define DHc  64
#define ENCc 32
#define KMIX 3
#define COUT 8

typedef float v2f __attribute__((ext_vector_type(2)));
typedef float v8f __attribute__((ext_vector_type(8)));

// ---------------- activations ----------------
template<int ACT>
__device__ __forceinline__ float actf(float x) {
  if (ACT == 1) return 0.5f * x * (1.0f + erff(x * 0.70710678118654752440f)); // exact GELU
  if (ACT == 2) return fmaxf(x, 0.0f);                                        // ReLU
  if (ACT == 3) return 1.0f / (1.0f + expf(-x));                              // sigmoid
  if (ACT == 4) return (x > 20.0f) ? x : log1pf(expf(x));                     // softplus
  return x;
}

#define WMMA_F32(a, b, c) __builtin_amdgcn_wmma_f32_16x16x4_f32(false, (a), false, (b), (short)0, (c), false, false)

// ---------------- WMMA f32 GEMM:  out = act(A @ W^T + bias) ----------------
// A: (M,K) row-major.  W row n at W[n*ldw + koff + k] (supports split ew_w0).
// One wave computes a 32x32 macro-tile (2x2 WMMA 16x16 tiles, 32 accum VGPRs):
// 4 b64 loads/lane feed 4 WMMAs per K-step -> 8 FLOP/byte from cache.
template<int ACT>
__global__ void wmma_gemm(const float* __restrict__ A, const float* __restrict__ W,
                          const float* __restrict__ bias, float* __restrict__ out,
                          int M, int N, int K, int ldw, int koff) {
  const int lane = threadIdx.x & 31;
  const int wave = threadIdx.x >> 5;
  const int n0   = blockIdx.x * 32;
  const int m0   = blockIdx.y * 128 + wave * 32;
  if (m0 >= M) return;                       // wave-uniform: EXEC stays all-1s for WMMA

  const int lr = lane & 15;
  int ar0 = m0 + lr;       if (ar0 >= M) ar0 = M - 1;   // clamp (C rows are independent)
  int ar1 = m0 + 16 + lr;  if (ar1 >= M) ar1 = M - 1;
  const int c0 = n0 + lr;        const bool ok0 = (c0 < N);
  const int c1 = n0 + 16 + lr;   const bool ok1 = (c1 < N);
  const int bc0 = ok0 ? c0 : (N - 1);
  const int bc1 = ok1 ? c1 : (N - 1);

  const float* __restrict__ Ap0 = A + (long)ar0 * K;
  const float* __restrict__ Ap1 = A + (long)ar1 * K;
  const float* __restrict__ Wp0 = W + (long)bc0 * ldw + koff;
  const float* __restrict__ Wp1 = W + (long)bc1 * ldw + koff;
  const int k2 = (lane >> 4) * 2;            // K offset of this half-wave's fragment pair

  v8f c00, c01, c10, c11;
  #pragma unroll
  for (int v = 0; v < 8; ++v) { c00[v] = 0.f; c01[v] = 0.f; c10[v] = 0.f; c11[v] = 0.f; }

  #pragma unroll 2
  for (int k = 0; k < K; k += 4) {
    v2f a0, a1, b0, b1;
    a0[0] = Ap0[k + k2];  a0[1] = Ap0[k + k2 + 1];
    a1[0] = Ap1[k + k2];  a1[1] = Ap1[k + k2 + 1];
    b0[0] = Wp0[k + k2];  b0[1] = Wp0[k + k2 + 1];
    b1[0] = Wp1[k + k2];  b1[1] = Wp1[k + k2 + 1];
    c00 = WMMA_F32(a0, b0, c00);
    c01 = WMMA_F32(a0, b1, c01);
    c10 = WMMA_F32(a1, b0, c10);
    c11 = WMMA_F32(a1, b1, c11);
  }

  const float bv0 = (bias != nullptr && ok0) ? bias[c0] : 0.0f;
  const float bv1 = (bias != nullptr && ok1) ? bias[c1] : 0.0f;
  const int rb0 = m0 + ((lane >> 4) << 3);   // C layout: VGPR v -> row v (+8 for hi lanes)
  #pragma unroll
  for (int v = 0; v < 8; ++v) {
    const int r0 = rb0 + v;
    if (r0 < M) {
      if (ok0) out[(long)r0 * N + c0] = actf<ACT>(c00[v] + bv0);
      if (ok1) out[(long)r0 * N + c1] = actf<ACT>(c01[v] + bv1);
    }
    const int r1 = rb0 + 16 + v;
    if (r1 < M) {
      if (ok0) out[(long)r1 * N + c0] = actf<ACT>(c10[v] + bv0);
      if (ok1) out[(long)r1 * N + c1] = actf<ACT>(c11[v] + bv1);
    }
  }
}

// ---------------- LayerNorm (opt residual, opt post-GELU). block=256, one token ----------------
template<bool RES, bool GELU_OUT>
__global__ void addln_kernel(const float* __restrict__ x, const float* __restrict__ r,
                             const float* __restrict__ g, const float* __restrict__ be,
                             float* __restrict__ out) {
  const int tok = blockIdx.x;
  const int t   = threadIdx.x;
  __shared__ float s1[256];
  __shared__ float s2[256];
  const float* xr = x + (long)tok * Dc;
  float x0 = xr[t], x1 = xr[t + 256];
  if (RES) { const float* rr = r + (long)tok * Dc; x0 += rr[t]; x1 += rr[t + 256]; }
  s1[t] = x0 + x1;
  s2[t] = x0 * x0 + x1 * x1;
  __syncthreads();
  for (int s = 128; s > 0; s >>= 1) {
    if (t < s) { s1[t] += s1[t + s]; s2[t] += s2[t + s]; }
    __syncthreads();
  }
  const float mean = s1[0] * (1.0f / Dc);
  const float var  = s2[0] * (1.0f / Dc) - mean * mean;
  const float rstd = rsqrtf(var + 1e-5f);
  float y0 = (x0 - mean) * rstd * g[t]       + be[t];
  float y1 = (x1 - mean) * rstd * g[t + 256] + be[t + 256];
  if (GELU_OUT) { y0 = actf<1>(y0); y1 = actf<1>(y1); }
  out[(long)tok * Dc + t]       = y0;
  out[(long)tok * Dc + t + 256] = y1;
}

// ---------------- sinusoidal PE add (in place) ----------------
__global__ void pe_add_kernel(float* __restrict__ h) {
  const long idx = (long)blockIdx.x * blockDim.x + threadIdx.x;
  if (idx >= (long)Bc * Lc * Dc) return;
  const int d   = (int)(idx % Dc);
  const int l   = (int)((idx / Dc) % Lc);
  const int i   = d >> 1;
  const float freq = expf((float)(2 * i) * (-logf(10000.0f) / (float)Dc));
  const float ang  = (float)l * freq;
  h[idx] += (d & 1) ? cosf(ang) : sinf(ang);
}

// ---------------- t *= sigmoid-scale (per token broadcast) ----------------
__global__ void rowscale_kernel(float* __restrict__ t, const float* __restrict__ sc) {
  const long idx = (long)blockIdx.x * blockDim.x + threadIdx.x;
  if (idx >= (long)Bc * Lc * Dc) return;
  t[idx] *= sc[idx / Dc];
}

// ---------------- attention: one block per (b,h,query). block=128 ----------------
__global__ void attn_kernel(const float* __restrict__ q, const float* __restrict__ k,
                            const float* __restrict__ v, float* __restrict__ o) {
  const int bid = blockIdx.x;               // b*H*L + h*L + qi
  const int qi  = bid % Lc;
  const int h   = (bid / Lc) % Hc;
  const int b   = bid / (Lc * Hc);
  __shared__ float sc[Lc];
  __shared__ float red[Lc];
  const int t = threadIdx.x;                // 0..127 == key index
  const float* qrow = q + ((long)(b * Lc + qi)) * Dc + h * DHc;
  const float* krow = k + ((long)(b * Lc + t)) * Dc + h * DHc;
  float acc = 0.0f;
  #pragma unroll 8
  for (int d = 0; d < DHc; ++d) acc += qrow[d] * krow[d];
  acc *= 0.125f;                            // 1/sqrt(64)
  sc[t] = acc; red[t] = acc;
  __syncthreads();
  for (int s = 64; s > 0; s >>= 1) { if (t < s) red[t] = fmaxf(red[t], red[t + s]); __syncthreads(); }
  const float m = red[0];
  __syncthreads();
  const float e = expf(sc[t] - m);
  sc[t] = e; red[t] = e;
  __syncthreads();
  for (int s = 64; s > 0; s >>= 1) { if (t < s) red[t] += red[t + s]; __syncthreads(); }
  const float inv = 1.0f / red[0];
  __syncthreads();
  sc[t] = sc[t] * inv;
  __syncthreads();
  if (t < DHc) {
    float a2 = 0.0f;
    for (int j = 0; j < Lc; ++j) a2 += sc[j] * v[((long)(b * Lc + j)) * Dc + h * DHc + t];
    o[((long)(b * Lc + qi)) * Dc + h * DHc + t] = a2;
  }
}

// ---------------- fused edge-weight MLP: ew[b,h,i,j] = sig( relu(l_i+r_j+b0) . w1_h + b1_h )
// one block per (b,i), 128 threads = one j each; never materializes the (B,L,L,D) tensor.
__global__ void edge_kernel(const float* __restrict__ left, const float* __restrict__ right,
                            const float* __restrict__ b0, const float* __restrict__ w1,
                            const float* __restrict__ b1, float* __restrict__ ew) {
  const int bid = blockIdx.x;               // b*L + i
  const int i = bid % Lc, b = bid / Lc;
  __shared__ float sL[Dc];
  __shared__ float sW[Hc * Dc];
  const int t = threadIdx.x;                // 0..127
  for (int idx = t; idx < Dc; idx += 128)      sL[idx] = left[(long)bid * Dc + idx] + b0[idx];
  for (int idx = t; idx < Hc * Dc; idx += 128) sW[idx] = w1[idx];
  __syncthreads();
  const int j = t;
  const float* rrow = right + ((long)(b * Lc + j)) * Dc;
  float acc[Hc];
  #pragma unroll
  for (int h = 0; h < Hc; ++h) acc[h] = 0.0f;
  for (int kx = 0; kx < Dc; ++kx) {
    const float p = fmaxf(sL[kx] + rrow[kx], 0.0f);
    #pragma unroll
    for (int h = 0; h < Hc; ++h) acc[h] += p * sW[h * Dc + kx];
  }
  #pragma unroll
  for (int h = 0; h < Hc; ++h) {
    const float val = 1.0f / (1.0f + expf(-(acc[h] + b1[h])));
    ew[(((long)(b * Hc + h)) * Lc + i) * Lc + j] = val;
  }
}

// ---------------- message passing: h5 = h4 + (ew @ xh)/L  (per-head) ----------------
__global__ void msg_kernel(const float* __restrict__ ew, const float* __restrict__ h4,
                           float* __restrict__ h5) {
  const int bid = blockIdx.x;               // b*L + i
  const int i = bid % Lc, b = bid / Lc;
  const int t = threadIdx.x;                // 0..255
  for (int d = t; d < Dc; d += 256) {
    const int h = d >> 6;                   // DH = 64
    const float* erow = ew + (((long)(b * Hc + h)) * Lc + i) * Lc;
    float acc = 0.0f;
    for (int j = 0; j < Lc; ++j) acc += erow[j] * h4[((long)(b * Lc + j)) * Dc + d];
    h5[(long)bid * Dc + d] = h4[(long)bid * Dc + d] + acc * (1.0f / Lc);
  }
}

// ---------------- mean pool over L ----------------
__global__ void pool_kernel(const float* __restrict__ h5, float* __restrict__ pooled) {
  const int idx = blockIdx.x * blockDim.x + threadIdx.x;
  if (idx >= Bc * Dc) return;
  const int b = idx / Dc, d = idx % Dc;
  float s = 0.0f;
  for (int l = 0; l < Lc; ++l) s += h5[((long)(b * Lc + l)) * Dc + d];
  pooled[idx] = s * (1.0f / Lc);
}

// ---------------- softmax over K=3 mixture logits ----------------
__global__ void softmax3_kernel(const float* __restrict__ x, float* __restrict__ out) {
  const int b = threadIdx.x;
  if (b >= Bc) return;
  const float a0 = x[b * 3 + 0], a1 = x[b * 3 + 1], a2 = x[b * 3 + 2];
  const float m  = fmaxf(a0, fmaxf(a1, a2));
  const float e0 = expf(a0 - m), e1 = expf(a1 - m), e2 = expf(a2 - m);
  const float inv = 1.0f / (e0 + e1 + e2);
  out[b * 3 + 0] = e0 * inv; out[b * 3 + 1] = e1 * inv; out[b * 3 + 2] = e2 * inv;
}

// ---------------- host-side dispatch ----------------
static void gemm(int act, const float* A, const float* W, const float* bias, float* out,
                 int M, int N, int K, int ldw, int koff, hipStream_t s) {
  dim3 grid((N + 31) / 32, (M + 127) / 128), block(128);
  switch (act) {
    case 1: wmma_gemm<1><<<grid, block, 0, s>>>(A, W, bias, out, M, N, K, ldw, koff); break;
    case 2: wmma_gemm<2><<<grid, block, 0, s>>>(A, W, bias, out, M, N, K, ldw, koff); break;
    case 3: wmma_gemm<3><<<grid, block, 0, s>>>(A, W, bias, out, M, N, K, ldw, koff); break;
    case 4: wmma_gemm<4><<<grid, block, 0, s>>>(A, W, bias, out, M, N, K, ldw, koff); break;
    default: wmma_gemm<0><<<grid, block, 0, s>>>(A, W, bias, out, M, N, K, ldw, koff); break;
  }
}

// input indices (setup_inputs dict order: x_enc then params in insertion order)
enum {
  IN_X = 0,
  EMB_W0, EMB_B0, EMB_G0, EMB_BE0, EMB_W1, EMB_B1, EMB_G1, EMB_BE1,
  N1_G, N1_B, N2_G, N2_B, N3_G, N3_B, N4_G, N4_B,
  SP_W0, SP_B0, SP_W1, SP_B1, SP_W2, SP_B2,
  AS_W0, AS_B0, AS_W1, AS_B1,
  TA_WQ, TA_WK, TA_WV, TA_WO, TA_BQ, TA_BK, TA_BV, TA_BO,
  GA_WQ, GA_WK, GA_WV, GA_WO, GA_BQ, GA_BK, GA_BV, GA_BO,
  EW_W0, EW_B0, EW_W1, EW_B1,
  PI_W0, PI_B0, PI_W1, PI_B1,
  MU_W0, MU_B0, MU_W1, MU_B1,
  SG_W0, SG_B0, SG_W1, SG_B1
};

extern "C" void kernel_launch(void* const* d_in, const int* in_sizes, int n_in,
                              void* d_out, int out_size, void* d_ws, size_t ws_size,
                              hipStream_t stream) {
  (void)in_sizes; (void)n_in; (void)out_size; (void)ws_size;
  #define P(i) ((const float*)d_in[(i)])
  float* out = (float*)d_out;
  float* ws  = (float*)d_ws;

  const long BLD  = (long)Bc * Lc * Dc;      // 524288
  const long BL2D = 2 * BLD;
  float* buf0 = ws;
  float* buf1 = ws + 1 * BLD;
  float* buf2 = ws + 2 * BLD;
  float* buf3 = ws + 3 * BLD;
  float* buf4 = ws + 4 * BLD;
  float* big0 = ws + 5 * BLD;                // (B,L,2D)
  float* big1 = ws + 5 * BLD + BL2D;         // (B,L,2D)
  float* ewb  = ws + 9 * BLD;                // (B,H,L,L) = BL2D floats
  float* as0  = ws + 11 * BLD;               // (B,L,128)
  float* scl  = as0 + (long)Bc * Lc * 128;   // (B,L)
  float* pooled = scl + Bc * Lc;             // (B,D)
  float* ht0  = pooled + Bc * Dc;            // (8,256)
  float* ht1  = ht0 + 8 * 256;               // (8,3)

  const int TOK = Bc * Lc;                   // 1024
  dim3 b256(256);
  const int ELT_GRID = (int)((BLD + 255) / 256);

  // ---- embedding: Linear -> LN -> GELU -> Linear -> LN, then extra LN(n1) ----
  gemm(0, P(IN_X), P(EMB_W0), P(EMB_B0), buf0, TOK, Dc, ENCc, ENCc, 0, stream);
  addln_kernel<false, true ><<<TOK, b256, 0, stream>>>(buf0, nullptr, P(EMB_G0), P(EMB_BE0), buf1);
  gemm(0, buf1, P(EMB_W1), P(EMB_B1), buf0, TOK, Dc, Dc, Dc, 0, stream);
  addln_kernel<false, false><<<TOK, b256, 0, stream>>>(buf0, nullptr, P(EMB_G1), P(EMB_BE1), buf1);
  addln_kernel<false, false><<<TOK, b256, 0, stream>>>(buf1, nullptr, P(N1_G),   P(N1_B),   buf2);
  pe_add_kernel<<<ELT_GRID, b256, 0, stream>>>(buf2);                 // h = buf2

  // ---- spatial encoder (D->2D->2D->D, GELU) + residual LN -> h2 = buf3 ----
  gemm(1, buf2, P(SP_W0), P(SP_B0), big0, TOK, 2 * Dc, Dc,     Dc,     0, stream);
  gemm(1, big0, P(SP_W1), P(SP_B1), big1, TOK, 2 * Dc, 2 * Dc, 2 * Dc, 0, stream);
  gemm(0, big1, P(SP_W2), P(SP_B2), buf0, TOK, Dc,     2 * Dc, 2 * Dc, 0, stream);
  addln_kernel<true, false><<<TOK, b256, 0, stream>>>(buf0, buf2, P(N2_G), P(N2_B), buf3);

  // ---- adaptive scale: sigmoid(W1 relu(W0 h2)) ----
  gemm(2, buf3, P(AS_W0), P(AS_B0), as0, TOK, Dc / 4, Dc, Dc, 0, stream);
  gemm(3, as0,  P(AS_W1), P(AS_B1), scl, TOK, 1, Dc / 4, Dc / 4, 0, stream);

  // ---- temporal attention ----
  gemm(0, buf3, P(TA_WQ), P(TA_BQ), buf0, TOK, Dc, Dc, Dc, 0, stream);
  gemm(0, buf3, P(TA_WK), P(TA_BK), buf1, TOK, Dc, Dc, Dc, 0, stream);
  gemm(0, buf3, P(TA_WV), P(TA_BV), buf2, TOK, Dc, Dc, Dc, 0, stream);
  attn_kernel<<<Bc * Hc * Lc, 128, 0, stream>>>(buf0, buf1, buf2, buf4);
  gemm(0, buf4, P(TA_WO), P(TA_BO), buf0, TOK, Dc, Dc, Dc, 0, stream);
  rowscale_kernel<<<ELT_GRID, b256, 0, stream>>>(buf0, scl);
  addln_kernel<true, false><<<TOK, b256, 0, stream>>>(buf0, buf3, P(N3_G), P(N3_B), buf4); // h3 = buf4

  // ---- graph attention ----
  gemm(0, buf4, P(GA_WQ), P(GA_BQ), buf0, TOK, Dc, Dc, Dc, 0, stream);
  gemm(0, buf4, P(GA_WK), P(GA_BK), buf1, TOK, Dc, Dc, Dc, 0, stream);
  gemm(0, buf4, P(GA_WV), P(GA_BV), buf2, TOK, Dc, Dc, Dc, 0, stream);
  attn_kernel<<<Bc * Hc * Lc, 128, 0, stream>>>(buf0, buf1, buf2, buf3);
  gemm(0, buf3, P(GA_WO), P(GA_BO), buf0, TOK, Dc, Dc, Dc, 0, stream);
  addln_kernel<true, false><<<TOK, b256, 0, stream>>>(buf0, buf4, P(N4_G), P(N4_B), buf2); // h4 = buf2

  // ---- dynamic edge weights + message passing ----
  gemm(0, buf2, P(EW_W0), nullptr, buf0, TOK, Dc, Dc, 2 * Dc, 0,  stream);  // left
  gemm(0, buf2, P(EW_W0), nullptr, buf1, TOK, Dc, Dc, 2 * Dc, Dc, stream);  // right
  edge_kernel<<<Bc * Lc, 128, 0, stream>>>(buf0, buf1, P(EW_B0), P(EW_W1), P(EW_B1), ewb);
  msg_kernel<<<Bc * Lc, b256, 0, stream>>>(ewb, buf2, buf3);                 // h5 = buf3

  // ---- pool + MDN heads ----
  pool_kernel<<<(Bc * Dc + 255) / 256, b256, 0, stream>>>(buf3, pooled);
  gemm(2, pooled, P(PI_W0), P(PI_B0), ht0, Bc, Dc / 2, Dc, Dc, 0, stream);
  gemm(0, ht0,    P(PI_W1), P(PI_B1), ht1, Bc, KMIX, Dc / 2, Dc / 2, 0, stream);
  softmax3_kernel<<<1, 32, 0, stream>>>(ht1, out);                           // pi -> out[0:24]
  gemm(2, pooled, P(MU_W0), P(MU_B0), ht0, Bc, Dc / 2, Dc, Dc, 0, stream);
  gemm(0, ht0,    P(MU_W1), P(MU_B1), out + 24, Bc, KMIX * COUT, Dc / 2, Dc / 2, 0, stream);
  gemm(2, pooled, P(SG_W0), P(SG_B0), ht0, Bc, Dc / 2, Dc, Dc, 0, stream);
  gemm(4, ht0,    P(SG_W1), P(SG_B1), out + 216, Bc, KMIX * COUT, Dc / 2, Dc / 2, 0, stream);
  #undef P
}